// ChamferDist_45157286150477
// MI455X (gfx1250) — compile-verified
//
#include <hip/hip_runtime.h>

typedef __attribute__((ext_vector_type(2))) float v2f;
typedef __attribute__((ext_vector_type(8))) float v8f;

#define WAVES_PER_BLOCK 4
#define TILES_PER_WAVE  4                                        // 4 x 16 adv columns per wave
#define ADV_PER_BLOCK   (WAVES_PER_BLOCK * TILES_PER_WAVE * 16)  // 256

// 3-input float min without clang's canonicalization overhead.
__device__ __forceinline__ float min3f(float a, float b, float c) {
  float r;
  asm("v_min3_num_f32 %0, %1, %2, %3" : "=v"(r) : "v"(a), "v"(b), "v"(c));
  return r;
}

// One-directional (adv2ori) Chamfer distance via V_WMMA_F32_16X16X4_F32.
// d(i,j) = |o_i|^2 + |a_j|^2 - 2 o_i . a_j
// A_i = (ox, oy, oz, |o|^2), B_j = (-2ax, -2ay, -2az, 1)  =>  D = |o_i|^2 - 2 o.a
// min_i D(i,j) + |a_j|^2 is the per-adv-point nearest distance.
__global__ __launch_bounds__(WAVES_PER_BLOCK * 32) void chamfer_adv2ori_wmma(
    const float* __restrict__ adv, const float* __restrict__ ori,
    const float* __restrict__ wts, float* __restrict__ out,
    int B, int N) {
  const int blocksPerBatch = N / ADV_PER_BLOCK;
  const int b    = blockIdx.x / blocksPerBatch;
  const int bb   = blockIdx.x % blocksPerBatch;
  const int wave = threadIdx.x >> 5;
  const int lane = threadIdx.x & 31;
  const int col  = lane & 15;
  const bool hi  = lane >= 16;

  const float* __restrict__ advb = adv + (size_t)b * N * 3;
  const float* __restrict__ orib = ori + (size_t)b * N * 3;

  const int advbase = bb * ADV_PER_BLOCK + wave * (TILES_PER_WAVE * 16);

  // ---- Build B matrices (4x16, K x N) for this wave's 64 adv points ----
  // 32-bit B layout: lane holds column N = lane%16; lanes 0-15 carry K=0 (v0), K=1 (v1),
  // lanes 16-31 carry K=2 (v0), K=3 (v1).
  v2f  Bm[TILES_PER_WAVE];
  float an[TILES_PER_WAVE];
#pragma unroll
  for (int t = 0; t < TILES_PER_WAVE; ++t) {
    const int j = advbase + t * 16 + col;
    const float ax = advb[3 * j + 0];
    const float ay = advb[3 * j + 1];
    const float az = advb[3 * j + 2];
    an[t]   = ax * ax + ay * ay + az * az;
    Bm[t].x = hi ? (-2.0f * az) : (-2.0f * ax);
    Bm[t].y = hi ? 1.0f         : (-2.0f * ay);
  }

  // Per-lane running min of (|o_i|^2 - 2 o.a) over all ori rows seen by this lane.
  float acc[TILES_PER_WAVE];
#pragma unroll
  for (int t = 0; t < TILES_PER_WAVE; ++t) acc[t] = 3.0e38f;

  // ---- Sweep all ori points, two 16-row tiles per iteration.
  // Both b96 loads issue back-to-back; tile 0's compute waits only on load 0
  // (s_wait_loadcnt 1) so load 1 flies under the first 4-WMMA block.
  for (int i0 = 0; i0 < N; i0 += 32) {
    const float* __restrict__ p0 = orib + 3 * (i0 + col);
    const float* __restrict__ p1 = orib + 3 * (i0 + 16 + col);
    const float x0 = p0[0], y0 = p0[1], z0 = p0[2];
    const float x1 = p1[0], y1 = p1[1], z1 = p1[2];

    // 32-bit A layout (16x4): lane holds row M = lane%16; lanes 0-15: K0 (v0), K1 (v1);
    // lanes 16-31: K2 (v0), K3 (v1).
    {
      const float on = x0 * x0 + y0 * y0 + z0 * z0;
      v2f A;
      A.x = hi ? z0 : x0;
      A.y = hi ? on : y0;
      v8f d[TILES_PER_WAVE];
#pragma unroll
      for (int t = 0; t < TILES_PER_WAVE; ++t) {
        v8f c = {};
        d[t] = __builtin_amdgcn_wmma_f32_16x16x4_f32(
            false, A, false, Bm[t], (short)0, c, false, false);
      }
      // Keep all 4 D tiles live so the 4 WMMAs issue back-to-back (distinct regs).
      __builtin_amdgcn_sched_barrier(0);
#pragma unroll
      for (int t = 0; t < TILES_PER_WAVE; ++t) {
        const float m1 = min3f(d[t][0], d[t][1], d[t][2]);
        const float m2 = min3f(d[t][3], d[t][4], d[t][5]);
        const float m3 = min3f(d[t][6], d[t][7], acc[t]);
        acc[t] = min3f(m1, m2, m3);
      }
    }
    {
      const float on = x1 * x1 + y1 * y1 + z1 * z1;
      v2f A;
      A.x = hi ? z1 : x1;
      A.y = hi ? on : y1;
      v8f d[TILES_PER_WAVE];
#pragma unroll
      for (int t = 0; t < TILES_PER_WAVE; ++t) {
        v8f c = {};
        d[t] = __builtin_amdgcn_wmma_f32_16x16x4_f32(
            false, A, false, Bm[t], (short)0, c, false, false);
      }
      __builtin_amdgcn_sched_barrier(0);
#pragma unroll
      for (int t = 0; t < TILES_PER_WAVE; ++t) {
        const float m1 = min3f(d[t][0], d[t][1], d[t][2]);
        const float m2 = min3f(d[t][3], d[t][4], d[t][5]);
        const float m3 = min3f(d[t][6], d[t][7], acc[t]);
        acc[t] = min3f(m1, m2, m3);
      }
    }
  }

  // ---- Combine lane halves (rows 0-7 vs 8-15 coverage) and add |a_j|^2 back ----
  float partial = 0.0f;
#pragma unroll
  for (int t = 0; t < TILES_PER_WAVE; ++t) {
    float mn = acc[t];
    mn = fminf(mn, __shfl_xor(mn, 16, 32));
    partial += mn + an[t];   // identical value in both lane halves
  }
  if (hi) partial = 0.0f;    // each adv column appears in both halves; count once

  // ---- Wave sum over 16 active lanes, one atomic per wave ----
#pragma unroll
  for (int off = 16; off >= 1; off >>= 1)
    partial += __shfl_xor(partial, off, 32);
  if (lane == 0) {
    const float scale = wts[b] / ((float)B * (float)N);
    atomicAdd(out, partial * scale);
  }
}

extern "C" void kernel_launch(void* const* d_in, const int* in_sizes, int n_in,
                              void* d_out, int out_size, void* d_ws, size_t ws_size,
                              hipStream_t stream) {
  const float* adv = (const float*)d_in[0];
  const float* ori = (const float*)d_in[1];
  const float* wts = (const float*)d_in[2];
  float* out = (float*)d_out;

  const int B = in_sizes[2];                 // weights: one per batch
  const int N = in_sizes[0] / (B * 3);       // adv_pc flat = B*N*3

  hipMemsetAsync(out, 0, sizeof(float), stream);

  const int blocksPerBatch = N / ADV_PER_BLOCK;
  dim3 grid(B * blocksPerBatch);
  dim3 block(WAVES_PER_BLOCK * 32);
  hipLaunchKernelGGL(chamfer_adv2ori_wmma, grid, block, 0, stream,
                     adv, ori, wts, out, B, N);
}